// AlphaEntmax_25005299597368
// MI455X (gfx1250) — compile-verified
//
#include <hip/hip_runtime.h>

typedef __attribute__((ext_vector_type(2))) float v2f;
typedef __attribute__((ext_vector_type(8))) float v8f;
typedef __attribute__((ext_vector_type(4))) unsigned int v4u;
typedef __attribute__((ext_vector_type(8))) int v8i;
typedef __attribute__((ext_vector_type(4))) int v4i;

#define TOKENS 16384
#define E      64
#define H      8
#define HE     512

// ---------------------------------------------------------------------------
// f32 WMMA helper: D = A(16x4) * B(4x16) + C
// ---------------------------------------------------------------------------
__device__ __forceinline__ v8f wmma4(v2f a, v2f b, v8f c) {
    return __builtin_amdgcn_wmma_f32_16x16x4_f32(
        /*neg_a=*/false, a, /*neg_b=*/false, b,
        /*c_mod=*/(short)0, c, /*reuse_a=*/false, /*reuse_b=*/false);
}

// ---------------------------------------------------------------------------
// Kernel 1/3: OUT[T x N] = X[T x Kdim] * W^T + bias,  W is [N x Kdim] row-major.
// One wave per 16x16 output tile.
// ---------------------------------------------------------------------------
__global__ __launch_bounds__(256)
void ent_gemm_wmma(const float* __restrict__ X, const float* __restrict__ W,
                   const float* __restrict__ bias, float* __restrict__ OUT,
                   int Kdim, int Ndim) {
    const int wave = threadIdx.x >> 5;
    const int lane = threadIdx.x & 31;
    const int half = lane >> 4;
    const int r    = lane & 15;

    const int ntiles = Ndim >> 4;
    const int gw = blockIdx.x * 8 + wave;
    const int tm = gw / ntiles;
    const int tn = gw % ntiles;
    if (tm >= (TOKENS >> 4)) return;

    const float* xrow = X + (size_t)(tm * 16 + r) * Kdim;
    const float* wrow = W + (size_t)(tn * 16 + r) * Kdim;

    v8f acc = {};
    for (int k = 0; k < Kdim; k += 4) {
        v2f a = *(const v2f*)(xrow + k + 2 * half);
        v2f b = *(const v2f*)(wrow + k + 2 * half);
        acc = wmma4(a, b, acc);
    }

    const int   col  = tn * 16 + r;
    const float bb   = bias[col];
    const int   row0 = tm * 16 + half * 8;
#pragma unroll
    for (int v = 0; v < 8; ++v) {
        OUT[(size_t)(row0 + v) * Ndim + col] = acc[v] + bb;
    }
}

// ---------------------------------------------------------------------------
// TDM: DMA one 512-float row from global into LDS (per-wave instruction).
// D# built per CDNA5 ISA ch.8: count=1, data_size=4B, 1 row of 512 elements.
// ---------------------------------------------------------------------------
#if defined(__gfx1250__) && __has_builtin(__builtin_amdgcn_tensor_load_to_lds) && \
    __has_builtin(__builtin_amdgcn_s_wait_tensorcnt)
#define ENT_HAVE_TDM 1
__device__ __forceinline__ void tdm_load_row(void* lds_dst, const float* gsrc) {
    const unsigned long long ga = (unsigned long long)(uintptr_t)gsrc;
    const unsigned int lds = (unsigned int)(uintptr_t)lds_dst;  // low 32b = LDS offset

    v4u g0;
    g0.x = 1u;                                            // count=1, load, no gather
    g0.y = lds;                                           // lds_addr
    g0.z = (unsigned int)ga;                              // global_addr[31:0]
    g0.w = (unsigned int)((ga >> 32) & 0x1FFFFFFull)      // global_addr[56:32]
         | (2u << 30);                                    // type=2 ("image")

    v8i g1;
    g1[0] = 0x20000;          // data_size=2 (4 bytes), mask=0, no pad/iterate
    g1[1] = (int)(512u << 16);// tensor_dim0 = 512 (low 16 bits at bit 48)
    g1[2] = (int)(1u << 16);  // tensor_dim1 = 1
    g1[3] = (int)(512u << 16);// tile_dim0 = 512
    g1[4] = 1;                // tile_dim1 = 1, tile_dim2 = 0
    g1[5] = 512;              // tensor_dim0_stride = 512
    g1[6] = 0;
    g1[7] = 0;

    v4i gz = {0, 0, 0, 0};
#if __clang_major__ >= 23
    v8i gz8 = {0, 0, 0, 0, 0, 0, 0, 0};
    __builtin_amdgcn_tensor_load_to_lds(g0, g1, gz, gz, gz8, 0);
#else
    __builtin_amdgcn_tensor_load_to_lds(g0, g1, gz, gz, 0);
#endif
}
#else
#define ENT_HAVE_TDM 0
#endif

// ---------------------------------------------------------------------------
// Kernel 2: per-token scores (WMMA) + alpha-entmax bisection + att@V^T (WMMA).
// One 256-thread block (8 wave32) per token; everything lives in LDS.
// ---------------------------------------------------------------------------
__device__ __forceinline__ float ent_p(float z, float inv, bool sq) {
    if (z <= 0.0f) return 0.0f;
    if (sq) return z * z;                        // exact path for alpha = 1.5
    return exp2f(inv * log2f(fmaxf(z, 1e-30f))); // general alpha
}
__device__ __forceinline__ float wred_max(float v) {
#pragma unroll
    for (int m = 16; m; m >>= 1) v = fmaxf(v, __shfl_xor(v, m, 32));
    return v;
}
__device__ __forceinline__ float wred_sum(float v) {
#pragma unroll
    for (int m = 16; m; m >>= 1) v += __shfl_xor(v, m, 32);
    return v;
}

__global__ __launch_bounds__(256)
void ent_attn_entmax(const float* __restrict__ kbuf, const float* __restrict__ qbuf,
                     const float* __restrict__ vbuf, const float* __restrict__ alpha_p,
                     float* __restrict__ resbuf) {
    __shared__ float lq[HE];
    __shared__ float lk[HE];
    __shared__ float lv[HE];
    __shared__ float sXa[E * E];

    const int t    = blockIdx.x;
    const int tid  = threadIdx.x;
    const int wave = tid >> 5;
    const int lane = tid & 31;
    const int half = lane >> 4;
    const int r    = lane & 15;

#if ENT_HAVE_TDM
    if (tid == 0) {   // wave 0 issues 3 TDM descriptors; TENSORcnt tracks them
        tdm_load_row(lk, kbuf + (size_t)t * HE);
        tdm_load_row(lq, qbuf + (size_t)t * HE);
        tdm_load_row(lv, vbuf + (size_t)t * HE);
        __builtin_amdgcn_s_wait_tensorcnt(0);
    }
#else
    for (int i = tid; i < HE; i += 256) {
        lk[i] = kbuf[(size_t)t * HE + i];
        lq[i] = qbuf[(size_t)t * HE + i];
        lv[i] = vbuf[(size_t)t * HE + i];
    }
#endif
    __syncthreads();

    const float alpha = alpha_p[0];
    const float am1   = alpha - 1.0f;
    const float inv   = 1.0f / am1;
    const bool  sq    = (inv == 2.0f);
    const float scale = am1 * 0.125f;  // am1 / sqrt(64)

    // Xa[i][j] = scale * sum_h q[h,i]*k[h,j] :  (64x8)@(8x64) via WMMA,
    // 4x4 grid of 16x16 tiles, 2 k-steps each; 2 tiles per wave.
#pragma unroll 1
    for (int tile = wave; tile < 16; tile += 8) {
        const int ti = tile >> 2, tj = tile & 3;
        v8f acc = {};
#pragma unroll
        for (int k0 = 0; k0 < H; k0 += 4) {
            const int ka = k0 + 2 * half;
            v2f a, b;
            a.x = lq[ka * E + ti * 16 + r];        // A[i][h] = q[h][i]
            a.y = lq[(ka + 1) * E + ti * 16 + r];
            b.x = lk[ka * E + tj * 16 + r];        // B[h][j] = k[h][j]
            b.y = lk[(ka + 1) * E + tj * 16 + r];
            acc = wmma4(a, b, acc);
        }
        const int col  = tj * 16 + r;
        const int row0 = ti * 16 + half * 8;
#pragma unroll
        for (int v = 0; v < 8; ++v) sXa[(row0 + v) * E + col] = acc[v] * scale;
    }
    __syncthreads();

    // Bisection: one wave per row; lane covers columns (lane, lane+32).
    const float thi_c = exp2f(-6.0f * am1);  // (1/64)^am1
#pragma unroll 1
    for (int rr = 0; rr < 8; ++rr) {
        const int row = rr * 8 + wave;
        const float z0 = sXa[row * E + lane];
        const float z1 = sXa[row * E + lane + 32];

        const float mv     = wred_max(fmaxf(z0, z1));
        float tau_lo       = mv - 1.0f;
        const float tau_hi = mv - thi_c;
        const float f_lo   = wred_sum(ent_p(z0 - tau_lo, inv, sq) +
                                      ent_p(z1 - tau_lo, inv, sq)) - 1.0f;
        float dm    = tau_hi - tau_lo;
        float tau_m = tau_lo;
#pragma unroll 1
        for (int it = 0; it < 50; ++it) {
            dm *= 0.5f;
            tau_m = tau_lo + dm;
            const float fm = wred_sum(ent_p(z0 - tau_m, inv, sq) +
                                      ent_p(z1 - tau_m, inv, sq)) - 1.0f;
            if (fm * f_lo >= 0.0f) tau_lo = tau_m;  // uniform across wave
        }
        const float p0 = ent_p(z0 - tau_m, inv, sq);
        const float p1 = ent_p(z1 - tau_m, inv, sq);
        const float rs = 1.0f / wred_sum(p0 + p1);
        sXa[row * E + lane]      = p0 * rs;
        sXa[row * E + lane + 32] = p1 * rs;
    }
    __syncthreads();

    // res^T[i][h] = sum_j att[i][j] * v[h][j] : (64x64)@(64x16) via WMMA,
    // B columns 8..15 zero-padded (lane select keeps EXEC all-ones).
    if (wave < 4) {
        const int ti = wave;
        v8f acc = {};
#pragma unroll 1
        for (int k = 0; k < E; k += 4) {
            const int ka = k + 2 * half;
            v2f a = *(const v2f*)(&sXa[(ti * 16 + r) * E + ka]); // att[i][j]
            v2f b;
            b.x = (r < H) ? lv[r * E + ka]     : 0.0f;           // B[j][h]=v[h][j]
            b.y = (r < H) ? lv[r * E + ka + 1] : 0.0f;
            acc = wmma4(a, b, acc);
        }
        if (r < H) {  // store only real head columns; res[h][i] at o = h*64+i
#pragma unroll
            for (int v = 0; v < 8; ++v) {
                resbuf[(size_t)t * HE + r * E + (ti * 16 + v + half * 8)] = acc[v];
            }
        }
    }
}

// ---------------------------------------------------------------------------
extern "C" void kernel_launch(void* const* d_in, const int* in_sizes, int n_in,
                              void* d_out, int out_size, void* d_ws, size_t ws_size,
                              hipStream_t stream) {
    (void)in_sizes; (void)n_in; (void)out_size; (void)ws_size;

    const float* x     = (const float*)d_in[0];
    const float* alpha = (const float*)d_in[1];
    const float* Wk    = (const float*)d_in[2];
    const float* bk    = (const float*)d_in[3];
    const float* Wq    = (const float*)d_in[4];
    const float* bq    = (const float*)d_in[5];
    const float* Wv    = (const float*)d_in[6];
    const float* bv    = (const float*)d_in[7];
    const float* Wu    = (const float*)d_in[8];
    const float* bu    = (const float*)d_in[9];
    float* out = (float*)d_out;

    float* kbuf = (float*)d_ws;
    float* qbuf = kbuf + (size_t)TOKENS * HE;
    float* vbuf = qbuf + (size_t)TOKENS * HE;

    const int proj_blocks = (TOKENS / 16) * (HE / 16) / 8;
    ent_gemm_wmma<<<proj_blocks, 256, 0, stream>>>(x, Wk, bk, kbuf, E, HE);
    ent_gemm_wmma<<<proj_blocks, 256, 0, stream>>>(x, Wq, bq, qbuf, E, HE);
    ent_gemm_wmma<<<proj_blocks, 256, 0, stream>>>(x, Wv, bv, vbuf, E, HE);

    ent_attn_entmax<<<TOKENS, 256, 0, stream>>>(kbuf, qbuf, vbuf, alpha, qbuf);

    const int out_blocks = (TOKENS / 16) * (E / 16) / 8;
    ent_gemm_wmma<<<out_blocks, 256, 0, stream>>>(qbuf, Wu, bu, out, HE, E);
}